// Concentration_17901423690231
// MI455X (gfx1250) — compile-verified
//
#include <hip/hip_runtime.h>
#include <stdint.h>

// ---------------- problem constants (from reference) ----------------
constexpr int kD        = 512;     // feature dim
constexpr int kNEdges   = 8192;    // number of segments
constexpr int kTotalIdx = 262144;  // flat member count
constexpr int kSegB     = 16;      // segments per block (WMMA M dim)
constexpr int kChunk    = 12;      // members staged per buffer (multiple of 4)
constexpr int kThreads  = 256;     // 8 waves (wave32)
// async b128 issues per thread per chunk: 12*128/256 == 6 (used in s_wait_asynccnt)
static_assert(kChunk * (kD / 4) / kThreads == 6, "update s_wait_asynccnt literal");

typedef __attribute__((ext_vector_type(2))) float v2f;
typedef __attribute__((ext_vector_type(8))) float v8f;

__device__ __forceinline__ int lower_bound_ll(const long long* a, int n, long long v) {
  int lo = 0, hi = n;
  while (lo < hi) {
    int mid = (lo + hi) >> 1;
    if (a[mid] < v) lo = mid + 1; else hi = mid;
  }
  return lo;
}

// Generic pointer to LDS: low 32 bits are the LDS byte offset (aperture rule).
__device__ __forceinline__ uint32_t lds_off_u32(const void* p) {
  return (uint32_t)(uintptr_t)p;
}

__global__ __launch_bounds__(kThreads) void seg_mean_wmma(
    const float* __restrict__ X,
    const long long* __restrict__ flat_idx,
    const long long* __restrict__ seg_ids,
    float* __restrict__ out) {
  __shared__ __align__(16) float buf[2][kChunk * kD];  // 2 x 24 KB staging tiles
  __shared__ int seg_off[kSegB + 1];
  __shared__ int sm_seg[2][kChunk];
  __shared__ int sm_row[2][kChunk];

  const int tid  = threadIdx.x;
  const int lane = tid & 31;
  const int wave = tid >> 5;
  const long long s0 = (long long)blockIdx.x * kSegB;

  // Segment ranges for this block's 16 segments (sorted segment_ids).
  if (tid <= kSegB) {
    seg_off[tid] = lower_bound_ll(seg_ids, kTotalIdx, s0 + tid);
  }
  __syncthreads();
  const int base  = seg_off[0];
  const int total = seg_off[kSegB] - base;

  v8f acc[4] = {};                 // 4 column tiles of 16 per wave (64 cols/wave)

  const int colbase = wave * 64;
  const int nIdx    = lane & 15;   // N (column) / M (segment) index within tile
  const int hi      = (lane >= 16) ? 1 : 0;

  // ---- pipeline helpers ----
  auto load_meta = [&](int c, int bsel) {
    if (tid < kChunk) {
      int g = c * kChunk + tid;
      int row = 0, sid = -1;       // tail: clamp to row 0, indicator never matches
      if (g < total) {
        row = (int)flat_idx[base + g];
        sid = (int)(seg_ids[base + g] - s0);
      }
      sm_row[bsel][tid] = row;
      sm_seg[bsel][tid] = sid;
    }
  };

  auto stage = [&](int bsel) {
    // kChunk * (kD/4) = 1536 16-byte elements; exactly 6 per thread, no divergence.
    for (int e = tid; e < kChunk * (kD / 4); e += kThreads) {
      int j   = e >> 7;            // row within chunk (kD/4 == 128)
      int c16 = e & 127;           // 16-byte element within row
      const float* gp = X + (size_t)sm_row[bsel][j] * kD + (size_t)c16 * 4;
      uint32_t lp = lds_off_u32(&buf[bsel][j * kD + c16 * 4]);
      asm volatile("global_load_async_to_lds_b128 %0, %1, off"
                   :: "v"(lp), "v"(gp) : "memory");
    }
  };

  auto compute = [&](int bsel) {
    // One-hot block-sparse GEMM: 4 members per V_WMMA_F32_16X16X4_F32 step.
    // A (16x4): lanes 0-15 -> M=lane, K={0,1}; lanes 16-31 -> K={2,3}.
#pragma unroll
    for (int kb = 0; kb < kChunk; kb += 4) {
      int kk = kb + (hi ? 2 : 0);
      v2f a;
      a.x = (sm_seg[bsel][kk]     == nIdx) ? 1.0f : 0.0f;
      a.y = (sm_seg[bsel][kk + 1] == nIdx) ? 1.0f : 0.0f;
#pragma unroll
      for (int t = 0; t < 4; ++t) {
        int col = colbase + t * 16 + nIdx;
        v2f b;                      // B (4x16): dual layout of A
        b.x = buf[bsel][kk * kD + col];
        b.y = buf[bsel][(kk + 1) * kD + col];
        acc[t] = __builtin_amdgcn_wmma_f32_16x16x4_f32(
            false, a, false, b, (short)0, acc[t], false, false);
      }
    }
  };

  // ---- software pipeline: async DMA of chunk c+1 overlaps WMMA of chunk c ----
  const int nch = (total + kChunk - 1) / kChunk;
  if (nch > 0) {
    load_meta(0, 0);
    __syncthreads();
    stage(0);
  }
  for (int c = 0; c < nch; ++c) {
    const int cur = c & 1;
    if (c + 1 < nch) {
      __syncthreads();             // compute(c-1) done: slot cur^1 reusable
      load_meta(c + 1, cur ^ 1);
      __syncthreads();             // meta for c+1 visible to all stagers
      stage(cur ^ 1);
      // 6 loads just issued; in-order completion => chunk c fully resident.
      asm volatile("s_wait_asynccnt 6" ::: "memory");
    } else {
      asm volatile("s_wait_asynccnt 0" ::: "memory");
    }
    __syncthreads();               // every wave's chunk-c data landed
    compute(cur);
  }

  // D layout (16x16 f32): VGPR v -> M = v + 8*(lane>=16), N = lane&15.
  const int mb = hi ? 8 : 0;
#pragma unroll
  for (int v = 0; v < 8; ++v) {
    int M = v + mb;
    int cnt = seg_off[M + 1] - seg_off[M];
    float scale = 1.0f / (float)((cnt > 1) ? cnt : 1);
    size_t rowoff = ((size_t)s0 + M) * kD;
#pragma unroll
    for (int t = 0; t < 4; ++t) {
      out[rowoff + colbase + t * 16 + nIdx] = acc[t][v] * scale;
    }
  }
}

extern "C" void kernel_launch(void* const* d_in, const int* in_sizes, int n_in,
                              void* d_out, int out_size, void* d_ws, size_t ws_size,
                              hipStream_t stream) {
  (void)in_sizes; (void)n_in; (void)out_size; (void)d_ws; (void)ws_size;
  const float*     X        = (const float*)d_in[0];
  const long long* flat_idx = (const long long*)d_in[1];
  const long long* seg_ids  = (const long long*)d_in[2];
  float*           out      = (float*)d_out;

  dim3 grid(kNEdges / kSegB);   // 512 blocks, 16 segments each
  dim3 block(kThreads);         // 8 wave32 waves
  seg_mean_wmma<<<grid, block, 0, stream>>>(X, flat_idx, seg_ids, out);
}